// Program_46643344834574
// MI455X (gfx1250) — compile-verified
//
#include <hip/hip_runtime.h>

// ---------------------------------------------------------------------------
// Thermostat MLP simulation for MI455X (gfx1250, wave32).
// Per step: per-element MLP 2->32->32->1 via V_WMMA_F32_16X16X4_F32 (f32 WMMA,
// K accumulated over 8 chunks of 4), then state update, then a global stable
// partition (off-first) implemented as count -> exclusive scan -> scatter.
// 40 steps = 120 kernel launches on `stream` (graph-capture safe).
// ---------------------------------------------------------------------------

typedef __attribute__((ext_vector_type(2))) float v2f;
typedef __attribute__((ext_vector_type(8))) float v8f;

#define T_ON  66.0f
#define T_OFF 78.0f
#define NSTEPS 40

__device__ __forceinline__ float relu_f(float v) { return fmaxf(v, 0.0f); }

__device__ __forceinline__ v8f splat8(float v) {
    v8f r;
#pragma unroll
    for (int i = 0; i < 8; ++i) r[i] = v;
    return r;
}

__device__ __forceinline__ v8f wmma4(v2f a, v2f b, v8f c) {
    // D = A(16x4 f32) * B(4x16 f32) + C(16x16 f32)
    return __builtin_amdgcn_wmma_f32_16x16x4_f32(
        /*neg_a=*/false, a, /*neg_b=*/false, b,
        /*c_mod=*/(short)0, c, /*reuse_a=*/false, /*reuse_b=*/false);
}

// ---------------------------------------------------------------------------
// Kernel A: per-element MLP + thermostat update + per-block off-count.
// 256 threads = 8 waves; each wave owns 32 consecutive elements (two 16-row
// WMMA A-tiles). Writes PRE-permute x_new/on_new at the element's slot.
// ---------------------------------------------------------------------------
__global__ void __launch_bounds__(256) step_mlp_kernel(
    const float* __restrict__ input,
    const float* __restrict__ W1, const float* __restrict__ b1,
    const float* __restrict__ W2, const float* __restrict__ b2,
    const float* __restrict__ W3, const float* __restrict__ b3,
    const float* __restrict__ x_cur, const float* __restrict__ on_cur,
    float* __restrict__ x_tmp, float* __restrict__ on_tmp,
    unsigned* __restrict__ blk_cnt, int first)
{
    __shared__ float lw1[64];          // W1 (2x32)
    __shared__ float lb1[32];          // b1
    __shared__ float lw3[32];          // W3 (32x1)
    __shared__ float lds_lin[8][32];
    __shared__ float lds_x[8][32];
    __shared__ float lds_y[8][32];
    __shared__ unsigned s_cnt;

    const int tid  = threadIdx.x;
    const int lane = tid & 31;
    const int wave = tid >> 5;
    const int half = lane >> 4;   // 0: lanes 0-15, 1: lanes 16-31
    const int lm   = lane & 15;   // row-within-tile / N index
    const int s    = blockIdx.x * 256 + tid;

    if (tid < 64) lw1[tid] = W1[tid];
    if (tid < 32) { lb1[tid] = b1[tid]; lw3[tid] = W3[tid]; }
    if (tid == 0) s_cnt = 0u;

    const float lin_l = input[2 * s];
    const float x_l   = first ? input[2 * s + 1] : x_cur[s];
    const float on_l  = first ? 0.0f : on_cur[s];

    lds_lin[wave][lane] = lin_l;
    lds_x[wave][lane]   = x_l;
    __syncthreads();

    // --- W2 (32x32) into B-register layout, 2 N-tiles x 8 K-chunks.
    // B 4x16 f32 layout (mirrors documented A layout): lanes 0-15 carry
    // K = {4c, 4c+1} in .x/.y for column N=lm; lanes 16-31 carry K = {4c+2,4c+3}.
    v2f Bw[2][8];
#pragma unroll
    for (int c = 0; c < 8; ++c) {
        const int kb = 4 * c + 2 * half;
#pragma unroll
        for (int nt = 0; nt < 2; ++nt) {
            const int n = nt * 16 + lm;
            v2f b;
            b.x = W2[(kb + 0) * 32 + n];
            b.y = W2[(kb + 1) * 32 + n];
            Bw[nt][c] = b;
        }
    }

    // C-init with layer-2 bias (depends on N only).
    const float bias0 = b2[lm];
    const float bias1 = b2[16 + lm];
    v8f acc[2][2];
    acc[0][0] = splat8(bias0); acc[0][1] = splat8(bias1);
    acc[1][0] = splat8(bias0); acc[1][1] = splat8(bias1);

    // Per-row state for A generation: A-tile mt row lm == wave element mt*16+lm.
    const float lin_r0 = lds_lin[wave][lm],      x_r0 = lds_x[wave][lm];
    const float lin_r1 = lds_lin[wave][16 + lm], x_r1 = lds_x[wave][16 + lm];

#pragma unroll
    for (int c = 0; c < 8; ++c) {
        const int kb = 4 * c + 2 * half;   // this lane's K pair within chunk c
        const float w1a0 = lw1[kb],      w1a1 = lw1[kb + 1];       // W1[0][k]
        const float w1b0 = lw1[32 + kb], w1b1 = lw1[32 + kb + 1];  // W1[1][k]
        const float bb0  = lb1[kb],      bb1  = lb1[kb + 1];

        v2f A0, A1;  // layer-1 output in A-register layout
        A0.x = relu_f(fmaf(lin_r0, w1a0, fmaf(x_r0, w1b0, bb0)));
        A0.y = relu_f(fmaf(lin_r0, w1a1, fmaf(x_r0, w1b1, bb1)));
        A1.x = relu_f(fmaf(lin_r1, w1a0, fmaf(x_r1, w1b0, bb0)));
        A1.y = relu_f(fmaf(lin_r1, w1a1, fmaf(x_r1, w1b1, bb1)));

        acc[0][0] = wmma4(A0, Bw[0][c], acc[0][0]);
        acc[0][1] = wmma4(A0, Bw[1][c], acc[0][1]);
        acc[1][0] = wmma4(A1, Bw[0][c], acc[1][0]);
        acc[1][1] = wmma4(A1, Bw[1][c], acc[1][1]);
    }

    // --- Layer 3: y[M] = sum_N relu(h2[M][N]) * W3[N] + b3.
    // C layout: vgpr v, lane -> (M = v + 8*half, N = lm + 16*nt).
    const float w3a = lw3[lm];
    const float w3b = lw3[16 + lm];
    const float b3v = b3[0];
#pragma unroll
    for (int mt = 0; mt < 2; ++mt) {
        float t[8];
#pragma unroll
        for (int v = 0; v < 8; ++v)
            t[v] = fmaf(relu_f(acc[mt][0][v]), w3a, relu_f(acc[mt][1][v]) * w3b);
        // butterfly sum over the 16-lane group (xor masks stay within group)
#pragma unroll
        for (int m = 1; m < 16; m <<= 1)
#pragma unroll
            for (int v = 0; v < 8; ++v)
                t[v] += __shfl_xor(t[v], m, 32);
        if (lm == 0) {
#pragma unroll
            for (int v = 0; v < 8; ++v)
                lds_y[wave][mt * 16 + half * 8 + v] = t[v] + b3v;
        }
    }
    __syncthreads();

    // --- Thermostat update for this lane's element.
    const float plant = lds_y[wave][lane] * 10.0f;
    const bool  off   = (on_l <= 0.5f);
    const float x_new = off ? (x_l - plant) : (x_l - plant + 10.0f);
    const float on_new = off ? ((x_new <= T_ON) ? 1.0f : on_l)
                             : ((x_new >  T_OFF) ? 0.0f : on_l);
    x_tmp[s]  = x_new;
    on_tmp[s] = on_new;

    const unsigned long long bal = __ballot(off);
    if (lane == 0) atomicAdd(&s_cnt, (unsigned)__popcll(bal));
    __syncthreads();
    if (tid == 0) blk_cnt[blockIdx.x] = s_cnt;
}

// ---------------------------------------------------------------------------
// Kernel B: exclusive scan of per-block off-counts (single block).
// ---------------------------------------------------------------------------
__global__ void __launch_bounds__(1024) scan_kernel(
    const unsigned* __restrict__ cnt, unsigned* __restrict__ offs,
    unsigned* __restrict__ total, int n)
{
    __shared__ unsigned sc[1024];
    const int tid = threadIdx.x;
    const unsigned v = (tid < n) ? cnt[tid] : 0u;
    sc[tid] = v;
    __syncthreads();
#pragma unroll
    for (int d = 1; d < 1024; d <<= 1) {
        const unsigned t = (tid >= d) ? sc[tid - d] : 0u;
        __syncthreads();
        sc[tid] += t;
        __syncthreads();
    }
    const unsigned incl = sc[tid];
    if (tid < n) offs[tid] = incl - v;     // exclusive
    if (tid == 1023) total[0] = incl;      // grand total of off-elements
}

// ---------------------------------------------------------------------------
// Kernel C: stable-partition scatter. off elements keep order at the front,
// on elements keep order behind totalOff. Writes next state + traj[t].
// ---------------------------------------------------------------------------
__global__ void __launch_bounds__(256) scatter_kernel(
    const float* __restrict__ x_tmp, const float* __restrict__ on_tmp,
    const float* __restrict__ on_cur,
    float* __restrict__ x_nxt, float* __restrict__ on_nxt,
    const unsigned* __restrict__ offs, const unsigned* __restrict__ total,
    float* __restrict__ traj, int step, int B, int first)
{
    __shared__ unsigned sc[256];
    const int tid = threadIdx.x;
    const int s   = blockIdx.x * 256 + tid;
    const float on_old = first ? 0.0f : on_cur[s];   // flag from PRE-update isOn
    const unsigned off = (on_old <= 0.5f) ? 1u : 0u;
    sc[tid] = off;
    __syncthreads();
#pragma unroll
    for (int d = 1; d < 256; d <<= 1) {
        const unsigned t = (tid >= d) ? sc[tid - d] : 0u;
        __syncthreads();
        sc[tid] += t;
        __syncthreads();
    }
    const unsigned excl      = sc[tid] - off;               // off-rank in block
    const unsigned offBefore = offs[blockIdx.x] + excl;     // global off-rank
    const unsigned dest = off ? offBefore
                              : (total[0] + (unsigned)s - offBefore);
    const float xv = x_tmp[s];
    x_nxt[dest]  = xv;
    on_nxt[dest] = on_tmp[s];
    traj[(size_t)step * (size_t)B + dest] = xv;
}

// ---------------------------------------------------------------------------
extern "C" void kernel_launch(void* const* d_in, const int* in_sizes, int n_in,
                              void* d_out, int out_size, void* d_ws, size_t ws_size,
                              hipStream_t stream)
{
    const float* input = (const float*)d_in[0];
    const float* W1 = (const float*)d_in[1];
    const float* b1 = (const float*)d_in[2];
    const float* W2 = (const float*)d_in[3];
    const float* b2 = (const float*)d_in[4];
    const float* W3 = (const float*)d_in[5];
    const float* b3 = (const float*)d_in[6];

    const int B    = in_sizes[0] / 2;   // 262144
    const int nblk = B / 256;           // 1024

    // Workspace: 6 B-sized float arrays (~6 MB) + scan scratch.
    float* ws_f  = (float*)d_ws;
    float* xb[2] = { ws_f,                 ws_f + (size_t)B     };
    float* xP    =   ws_f + 2 * (size_t)B;
    float* ob[2] = { ws_f + 3 * (size_t)B, ws_f + 4 * (size_t)B };
    float* oP    =   ws_f + 5 * (size_t)B;
    unsigned* cnt  = (unsigned*)(ws_f + 6 * (size_t)B);
    unsigned* offs = cnt + nblk;
    unsigned* tot  = offs + nblk;

    float* traj = (float*)d_out;

    for (int t = 0; t < NSTEPS; ++t) {
        const int cur = t & 1, nxt = cur ^ 1;
        const int first = (t == 0) ? 1 : 0;
        step_mlp_kernel<<<nblk, 256, 0, stream>>>(
            input, W1, b1, W2, b2, W3, b3,
            xb[cur], ob[cur], xP, oP, cnt, first);
        scan_kernel<<<1, 1024, 0, stream>>>(cnt, offs, tot, nblk);
        scatter_kernel<<<nblk, 256, 0, stream>>>(
            xP, oP, ob[cur], xb[nxt], ob[nxt], offs, tot, traj, t, B, first);
    }
}